// AttentionFlow_57973468561807
// MI455X (gfx1250) — compile-verified
//
#include <hip/hip_runtime.h>

#define NEGF (-10000000.0f)

constexpr int Bb = 32;
constexpr int Lp = 1024;
constexpr int Lq = 128;
constexpr int Dd = 256;
constexpr int BM = 128;          // paragraph rows per workgroup
constexpr int QS = Dd + 2;       // padded LDS stride for Q (floats) == TDM pad
constexpr int AS = Lq + 2;       // padded LDS stride for att rows (floats)

typedef float v2f __attribute__((ext_vector_type(2)));
typedef float v8f __attribute__((ext_vector_type(8)));
typedef unsigned int u32x4 __attribute__((ext_vector_type(4)));
typedef int i32x4 __attribute__((ext_vector_type(4)));
typedef int i32x8 __attribute__((ext_vector_type(8)));

#if defined(__gfx1250__) && __has_builtin(__builtin_amdgcn_tensor_load_to_lds) && \
    __has_builtin(__builtin_amdgcn_s_wait_tensorcnt)
#define USE_TDM 1
#else
#define USE_TDM 0
#endif

__device__ __forceinline__ v8f zero8() {
    v8f z = {0.f, 0.f, 0.f, 0.f, 0.f, 0.f, 0.f, 0.f};
    return z;
}

// D = A(16x4 f32) * B(4x16 f32) + C(16x16 f32), full fp32 matrix path on CDNA5
__device__ __forceinline__ v8f wmma_f32(v2f a, v2f b, v8f c) {
    return __builtin_amdgcn_wmma_f32_16x16x4_f32(
        /*neg_a=*/false, a, /*neg_b=*/false, b,
        /*c_mod=*/(short)0, c, /*reuse_a=*/false, /*reuse_b=*/false);
}

// ---------------------------------------------------------------------------
// K1: per (batch, 128-row paragraph block):
//   S = (P . wx) + (Q . wy) + (P*wxy) Q^T + b   (masked by qm)
//   pq_att = softmax_j(S);  out1 = pq_att @ Q;  qp_sim = max_j(S)
// ---------------------------------------------------------------------------
extern "C" __global__ __launch_bounds__(256)
void af_k1(const float* __restrict__ P, const float* __restrict__ Q,
           const float* __restrict__ qm, const float* __restrict__ w,
           const float* __restrict__ bscal,
           float* __restrict__ out1, float* __restrict__ qp_sim)
{
    extern __shared__ float smem[];
    float* Qs   = smem;                      // Lq * QS   (LDS byte offset 0)
    float* atts = Qs + Lq * QS;              // 8 waves * 16 rows * AS
    float* wxs  = atts + 8 * 16 * AS;        // D
    float* wys  = wxs + Dd;                  // D
    float* wxys = wys + Dd;                  // D
    float* qms  = wxys + Dd;                 // Lq
    float* qys  = qms + Lq;                  // Lq
    float* pxs  = qys + Lq;                  // BM

    const int tid     = threadIdx.x;
    const int batch   = blockIdx.y;
    const int rowbase = blockIdx.x * BM;
    const float* Pb = P + ((size_t)batch * Lp) * Dd;
    const float* Qb = Q + ((size_t)batch * Lq) * Dd;

    const int wave = tid >> 5;
    const int lane = tid & 31;

#if USE_TDM
    // ---- Tensor Data Mover: DMA Q (128x256 f32) into LDS with 2-DWORD row
    // padding (pad_interval=7 -> every 256 DWORDs, pad_amount=1 -> 2 DWORDs),
    // i.e. exactly the QS=258 bank-conflict-free layout. Issued by wave 0,
    // tracked on TENSORcnt, published to the WG by the barrier below.
    if (wave == 0) {
        const unsigned long long ga = (unsigned long long)(const void*)Qb;
        u32x4 g0;
        g0.x = 1u;                                   // count=1 (valid), user D#
        g0.y = 0u;                                   // lds_addr = 0 (Qs)
        g0.z = (unsigned int)(ga & 0xFFFFFFFFull);   // global_addr[31:0]
        g0.w = (unsigned int)((ga >> 32) & 0x1FFFFFFull) | (2u << 30); // type=2
        i32x8 g1;
        g1[0] = (int)((2u << 16)     // data_size = 4 bytes
                    | (1u << 20)     // pad_enable
                    | (7u << 22)     // pad_interval: every 256 DWORDs
                    | (1u << 25));   // pad_amount: 2 DWORDs
        g1[1] = (int)(256u << 16);   // tensor_dim0[15:0]=256 (bits 63:48)
        g1[2] = (int)(128u << 16);   // tensor_dim1[15:0]=128 (bits 95:80)
        g1[3] = (int)(256u << 16);   // tile_dim0=256        (bits 127:112)
        g1[4] = 128;                 // tile_dim1=128        (bits 143:128)
        g1[5] = 256;                 // tensor_dim0_stride=256 (bits 191:160)
        g1[6] = 0;
        g1[7] = 0;
        i32x4 z4 = {0, 0, 0, 0};              // groups 2/3 unused (2-D tensor)
        i32x8 z8 = {0, 0, 0, 0, 0, 0, 0, 0};  // extra group (clang-23 form)
        __builtin_amdgcn_tensor_load_to_lds(g0, g1, z4, z4, z8, 0);
    }
#else
    for (int i = tid; i < Lq * Dd; i += 256) {
        const int r = i >> 8;            // Dd == 256
        const int c = i & (Dd - 1);
        Qs[r * QS + c] = Qb[i];
    }
#endif

    // ---- stage weights + mask (tiny; overlaps the TDM transfer) ----
    for (int i = tid; i < Dd; i += 256) {
        wxs[i]  = w[i];
        wys[i]  = w[Dd + i];
        wxys[i] = w[2 * Dd + i];
    }
    for (int i = tid; i < Lq; i += 256) qms[i] = qm[batch * Lq + i];

#if USE_TDM
    if (wave == 0) __builtin_amdgcn_s_wait_tensorcnt(0);
#endif
    __syncthreads();

    // ---- row/col bias terms ----
    if (tid < Lq) {
        float s = 0.f;
        for (int d = 0; d < Dd; ++d) s += Qs[tid * QS + d] * wys[d];
        qys[tid] = s;
    }
    if (tid < BM) {
        const float* pr = Pb + (size_t)(rowbase + tid) * Dd;
        float s = 0.f;
        for (int d = 0; d < Dd; ++d) s += pr[d] * wxs[d];
        pxs[tid] = s;
    }
    __syncthreads();

    const int half = lane >> 4;      // K-split half per ISA A/B layout
    const int lm   = lane & 15;      // M for A-frag, N for B-frag
    const int wrow = rowbase + wave * 16;

    // ---- GEMM1: S tile 16x128 = (P*wxy) @ Q^T, fp32 WMMA, K step 4 ----
    v8f acc[8];
    #pragma unroll
    for (int t = 0; t < 8; ++t) acc[t] = zero8();

    const float* prow = Pb + (size_t)(wrow + lm) * Dd;
    for (int k0 = 0; k0 < Dd; k0 += 4) {
        const int ka = k0 + 2 * half;
        v2f a;                                // A: lane=M, K = ka + r
        a.x = prow[ka]     * wxys[ka];
        a.y = prow[ka + 1] * wxys[ka + 1];
        #pragma unroll
        for (int t = 0; t < 8; ++t) {
            v2f bq;                           // B[d][j] = Q[j][d]; lane=N(j)
            bq.x = Qs[(t * 16 + lm) * QS + ka];
            bq.y = Qs[(t * 16 + lm) * QS + ka + 1];
            acc[t] = wmma_f32(a, bq, acc[t]);
        }
    }

    // ---- bias + mask + in-register softmax over j (128) ----
    const float bias = *bscal;
    float rmax[8], rsum[8];
    #pragma unroll
    for (int e = 0; e < 8; ++e) rmax[e] = NEGF;

    #pragma unroll
    for (int t = 0; t < 8; ++t) {
        const int j = t * 16 + lm;
        const float qb = qys[j] + bias;
        const bool on = qms[j] > 0.f;
        #pragma unroll
        for (int e = 0; e < 8; ++e) {
            const int row = e + 8 * half;
            float v = acc[t][e] + pxs[wave * 16 + row] + qb;
            v = on ? v : NEGF;
            acc[t][e] = v;
            rmax[e] = fmaxf(rmax[e], v);
        }
    }
    #pragma unroll
    for (int off = 1; off <= 8; off <<= 1) {
        #pragma unroll
        for (int e = 0; e < 8; ++e)
            rmax[e] = fmaxf(rmax[e], __shfl_xor(rmax[e], off, 32));
    }

    #pragma unroll
    for (int e = 0; e < 8; ++e) rsum[e] = 0.f;
    #pragma unroll
    for (int t = 0; t < 8; ++t) {
        #pragma unroll
        for (int e = 0; e < 8; ++e) {
            const float ev = __expf(acc[t][e] - rmax[e]);
            acc[t][e] = ev;
            rsum[e] += ev;
        }
    }
    #pragma unroll
    for (int off = 1; off <= 8; off <<= 1) {
        #pragma unroll
        for (int e = 0; e < 8; ++e)
            rsum[e] += __shfl_xor(rsum[e], off, 32);
    }

    // qp_sim = max_j(masked S)
    if (lm == 0) {
        #pragma unroll
        for (int e = 0; e < 8; ++e)
            qp_sim[(size_t)batch * Lp + wrow + e + 8 * half] = rmax[e];
    }

    // normalized attention: C-layout scatter -> LDS (A-layout gather below)
    #pragma unroll
    for (int t = 0; t < 8; ++t) {
        #pragma unroll
        for (int e = 0; e < 8; ++e) {
            const int row = e + 8 * half;
            atts[(wave * 16 + row) * AS + t * 16 + lm] = acc[t][e] / rsum[e];
        }
    }
    __syncthreads();

    // ---- GEMM2: out1 16x256 = att(16x128) @ Q(128x256) ----
    #pragma unroll 1
    for (int nt = 0; nt < 16; ++nt) {
        v8f c = zero8();
        for (int k0 = 0; k0 < Lq; k0 += 4) {
            const int ka = k0 + 2 * half;
            v2f a, bq;
            a.x  = atts[(wave * 16 + lm) * AS + ka];       // A: lane=M(i), K=j
            a.y  = atts[(wave * 16 + lm) * AS + ka + 1];
            bq.x = Qs[ka * QS + nt * 16 + lm];             // B[j][d]; lane=N(d)
            bq.y = Qs[(ka + 1) * QS + nt * 16 + lm];
            c = wmma_f32(a, bq, c);
        }
        #pragma unroll
        for (int e = 0; e < 8; ++e) {
            const int row = e + 8 * half;
            out1[((size_t)batch * Lp + wrow + row) * Dd + nt * 16 + lm] = c[e];
        }
    }
}

// ---------------------------------------------------------------------------
// K2: per batch: qp_att = softmax_i(where(dm>0, qp_sim, NEG));
//     qp_vec[d] = sum_i qp_att[i] * P[i][d]
// ---------------------------------------------------------------------------
extern "C" __global__ __launch_bounds__(256)
void af_k2(const float* __restrict__ P, const float* __restrict__ dm,
           const float* __restrict__ qp_sim, float* __restrict__ qp_vec)
{
    __shared__ float att[Lp];
    __shared__ float red[256];
    const int tid = threadIdx.x;
    const int batch = blockIdx.x;

    float lmax = NEGF;
    for (int i = tid; i < Lp; i += 256) {
        const float v = (dm[batch * Lp + i] > 0.f)
                      ? qp_sim[(size_t)batch * Lp + i] : NEGF;
        att[i] = v;
        lmax = fmaxf(lmax, v);
    }
    red[tid] = lmax; __syncthreads();
    for (int s = 128; s > 0; s >>= 1) {
        if (tid < s) red[tid] = fmaxf(red[tid], red[tid + s]);
        __syncthreads();
    }
    const float mx = red[0];
    __syncthreads();

    float lsum = 0.f;
    for (int i = tid; i < Lp; i += 256) {
        const float e = __expf(att[i] - mx);
        att[i] = e;
        lsum += e;
    }
    red[tid] = lsum; __syncthreads();
    for (int s = 128; s > 0; s >>= 1) {
        if (tid < s) red[tid] += red[tid + s];
        __syncthreads();
    }
    const float inv = 1.f / red[0];

    // GEMV: thread owns column d = tid (coalesced over i)
    float s = 0.f;
    const float* Pb = P + ((size_t)batch * Lp) * Dd + tid;
    for (int i = 0; i < Lp; ++i) s += att[i] * Pb[(size_t)i * Dd];
    qp_vec[batch * Dd + tid] = s * inv;
}

// ---------------------------------------------------------------------------
// K3: tiled_qp broadcast (wide grid so writes use full HBM bandwidth)
// ---------------------------------------------------------------------------
extern "C" __global__ __launch_bounds__(256)
void af_k3(const float* __restrict__ qp_vec, float* __restrict__ out2)
{
    const size_t idx = (size_t)blockIdx.x * 256 + threadIdx.x;
    const int b = (int)(idx / ((size_t)Lp * Dd));
    const int d = (int)(idx & (size_t)(Dd - 1));
    out2[idx] = qp_vec[b * Dd + d];
}

// ---------------------------------------------------------------------------
extern "C" void kernel_launch(void* const* d_in, const int* in_sizes, int n_in,
                              void* d_out, int out_size, void* d_ws, size_t ws_size,
                              hipStream_t stream)
{
    const float* P  = (const float*)d_in[0];   // paragraph (B,Lp,D)
    const float* Q  = (const float*)d_in[1];   // query     (B,Lq,D)
    const float* dm = (const float*)d_in[2];   // (B,Lp)
    const float* qm = (const float*)d_in[3];   // (B,Lq)
    const float* w  = (const float*)d_in[4];   // (3D,)
    const float* bs = (const float*)d_in[5];   // scalar

    float* out1 = (float*)d_out;                       // pq_vectors
    float* out2 = out1 + (size_t)Bb * Lp * Dd;         // tiled_qp

    float* qp_sim = (float*)d_ws;                      // B*Lp floats
    float* qp_vec = qp_sim + (size_t)Bb * Lp;          // B*D floats

    const size_t smem1 =
        (size_t)(Lq * QS + 8 * 16 * AS + 3 * Dd + 2 * Lq + BM) * sizeof(float);

    dim3 g1(Lp / BM, Bb);
    af_k1<<<g1, 256, smem1, stream>>>(P, Q, qm, w, bs, out1, qp_sim);
    af_k2<<<Bb, 256, 0, stream>>>(P, dm, qp_sim, qp_vec);

    const size_t tot = (size_t)Bb * Lp * Dd;
    af_k3<<<(unsigned)(tot / 256), 256, 0, stream>>>(qp_vec, out2);
}